// GMMLoss_pca_73547019977336
// MI455X (gfx1250) — compile-verified
//
#include <hip/hip_runtime.h>
#include <hip/hip_bf16.h>
#include <math.h>

#define NROWS 16384
#define NCOLS 1024
#define KDIM  16
#define NCLS  10
#define NCELL 2730    // 10 counts + 160 sums + 2560 second moments
#define NCHUNK 16
#define NCYCLE 32

typedef __attribute__((ext_vector_type(2))) float v2f;
typedef __attribute__((ext_vector_type(8))) float v8f;

// ---------------- column sums (deterministic chunked partials) ----------------
__global__ void colsum_part_kernel(const float* __restrict__ mu, float* __restrict__ part) {
    int gid = blockIdx.x * blockDim.x + threadIdx.x;     // 16384 threads
    int chunk = gid >> 10;
    int col   = gid & 1023;
    const float* p = mu + (size_t)chunk * 1024 * NCOLS + col;
    float acc = 0.f;
    for (int r = 0; r < 1024; ++r) { acc += *p; p += NCOLS; }
    part[chunk * NCOLS + col] = acc;
}

__global__ void colsum_reduce_kernel(const float* __restrict__ part, float* __restrict__ s) {
    int col = blockIdx.x * blockDim.x + threadIdx.x;
    if (col < NCOLS) {
        float a = 0.f;
        for (int c = 0; c < NCHUNK; ++c) a += part[c * NCOLS + col];
        s[col] = a;
    }
}

// ---------------- G = mu^T mu via V_WMMA_F32_16X16X4_F32 ----------------
// One wave per 32x32 super-tile (2x2 register blocking of 16x16 WMMA tiles):
// per K=4 step, 8 operand dwords/lane feed 4 WMMAs (2 dwords per WMMA vs 4
// for unblocked) -> halves L2 traffic, doubles arithmetic density.
// Grid = 32x32 = 1024 waves (~one per SIMD32 across the chip).
__global__ void gram_wmma_kernel(const float* __restrict__ mu, float* __restrict__ G) {
    int tile = blockIdx.x;               // 1024 super-tiles = 32 x 32
    int mt = tile >> 5, nt = tile & 31;
    int lane = threadIdx.x;
    int half = lane >> 4, lm = lane & 15;
    // A layout (32-bit 16x4): lanes 0-15 hold M=0..15; VGPR0: K = 2*half, VGPR1: K = 2*half+1
    const float* base = mu + (size_t)(2 * half) * NCOLS;
    const float* pa0 = base + mt * 32 + lm;
    const float* pa1 = pa0 + 16;
    const float* pb0 = base + nt * 32 + lm;
    const float* pb1 = pb0 + 16;
    v8f acc00 = {0.f, 0.f, 0.f, 0.f, 0.f, 0.f, 0.f, 0.f};
    v8f acc01 = acc00, acc10 = acc00, acc11 = acc00;
#pragma unroll 2
    for (int k = 0; k < NROWS; k += 4) {
        v2f a0, a1, b0, b1;
        a0.x = pa0[0]; a0.y = pa0[NCOLS];
        a1.x = pa1[0]; a1.y = pa1[NCOLS];
        b0.x = pb0[0]; b0.y = pb0[NCOLS];
        b1.x = pb1[0]; b1.y = pb1[NCOLS];
        acc00 = __builtin_amdgcn_wmma_f32_16x16x4_f32(false, a0, false, b0, (short)0, acc00, false, false);
        acc01 = __builtin_amdgcn_wmma_f32_16x16x4_f32(false, a0, false, b1, (short)0, acc01, false, false);
        acc10 = __builtin_amdgcn_wmma_f32_16x16x4_f32(false, a1, false, b0, (short)0, acc10, false, false);
        acc11 = __builtin_amdgcn_wmma_f32_16x16x4_f32(false, a1, false, b1, (short)0, acc11, false, false);
        pa0 += 4 * NCOLS; pa1 += 4 * NCOLS;
        pb0 += 4 * NCOLS; pb1 += 4 * NCOLS;
    }
    // C/D layout: VGPR v -> M = v + 8*half, N = lane&15
    int n0 = nt * 32 + lm;
    int m0 = mt * 32 + 8 * half;
#pragma unroll
    for (int v = 0; v < 8; ++v) {
        G[(size_t)(m0 + v) * NCOLS + n0]           = acc00[v];
        G[(size_t)(m0 + v) * NCOLS + n0 + 16]      = acc01[v];
        G[(size_t)(m0 + 16 + v) * NCOLS + n0]      = acc10[v];
        G[(size_t)(m0 + 16 + v) * NCOLS + n0 + 16] = acc11[v];
    }
}

// C = G - s s^T / B   (rank-1 centering correction, in place)
__global__ void center_kernel(float* __restrict__ C, const float* __restrict__ s) {
    size_t gid = (size_t)blockIdx.x * blockDim.x + threadIdx.x;   // 1M threads
    int i = (int)(gid >> 10), j = (int)(gid & 1023);
    C[gid] -= s[i] * s[j] * (1.0f / (float)NROWS);
}

// ---------------- subspace iteration ----------------
__global__ void initq_kernel(float* __restrict__ Q) {
    int gid = blockIdx.x * blockDim.x + threadIdx.x;   // 16384
    unsigned h = (unsigned)gid * 2654435761u;
    h ^= h >> 16; h *= 0x45d9f3bu; h ^= h >> 16;
    Q[gid] = (float)(h & 0xFFFFFFu) * (1.0f / 8388608.0f) - 1.0f;
}

// Z = C @ Q  (1024x1024 * 1024x16); thread per (row, n), n fastest for broadcast reads
__global__ void spmm_kernel(const float* __restrict__ C, const float* __restrict__ Qin,
                            float* __restrict__ Zout) {
    int t = threadIdx.x;
    int n = t & 15;
    int row = blockIdx.x * 16 + (t >> 4);
    const float* cr = C + (size_t)row * NCOLS;
    float acc = 0.f;
    for (int j = 0; j < NCOLS; ++j) acc += cr[j] * Qin[j * KDIM + n];
    Zout[row * KDIM + n] = acc;
}

// Cholesky-QR orthonormalization: T = Z^T Z, T = R^T R, Q = Z R^-1. Single WG.
__global__ void ortho_kernel(const float* __restrict__ Zin, float* __restrict__ Qout) {
    __shared__ float T[16][16];
    __shared__ float R[16][16];
    __shared__ float Rinv[16][16];
    int t = threadIdx.x;           // 256 threads
    int a = t >> 4, b = t & 15;
    float acc = 0.f;
    for (int j = 0; j < NCOLS; ++j) acc += Zin[j * KDIM + a] * Zin[j * KDIM + b];
    T[a][b] = acc;
    __syncthreads();
    if (t == 0) {
        for (int i = 0; i < 16; ++i) {
            float d = T[i][i];
            for (int k = 0; k < i; ++k) d -= R[k][i] * R[k][i];
            float rii = sqrtf(fmaxf(d, 1e-30f));
            R[i][i] = rii;
            float inv = 1.0f / rii;
            for (int j = i + 1; j < 16; ++j) {
                float v = T[i][j];
                for (int k = 0; k < i; ++k) v -= R[k][i] * R[k][j];
                R[i][j] = v * inv;
            }
        }
        for (int j = 0; j < 16; ++j) {           // solve R * Rinv = I (upper tri)
            for (int i = j; i >= 0; --i) {
                float v = (i == j) ? 1.f : 0.f;
                for (int k = i + 1; k <= j; ++k) v -= R[i][k] * Rinv[k][j];
                Rinv[i][j] = v / R[i][i];
            }
            for (int i = j + 1; i < 16; ++i) Rinv[i][j] = 0.f;
        }
    }
    __syncthreads();
    for (int rr = t; rr < NCOLS; rr += 256) {    // each row owned by one thread -> in-place safe
        float z[16];
        for (int n = 0; n < 16; ++n) z[n] = Zin[rr * KDIM + n];
        for (int n = 0; n < 16; ++n) {
            float v = 0.f;
            for (int m = 0; m <= n; ++m) v += z[m] * Rinv[m][n];
            Qout[rr * KDIM + n] = v;
        }
    }
}

// r[n] = (s/B) . Q[:,n]
__global__ void rvec_kernel(const float* __restrict__ s, const float* __restrict__ Q,
                            float* __restrict__ r) {
    int n = threadIdx.x;
    if (n < KDIM) {
        float acc = 0.f;
        for (int j = 0; j < NCOLS; ++j) acc += s[j] * Q[j * KDIM + n];
        r[n] = acc * (1.0f / (float)NROWS);
    }
}

// x = mu @ Q - r  (centered projection)
__global__ void project_kernel(const float* __restrict__ mu, const float* __restrict__ Q,
                               const float* __restrict__ r, float* __restrict__ x) {
    int t = threadIdx.x;
    int n = t & 15;
    int row = blockIdx.x * 16 + (t >> 4);
    const float* mr = mu + (size_t)row * NCOLS;
    float acc = -r[n];
    for (int j = 0; j < NCOLS; ++j) acc += mr[j] * Q[j * KDIM + n];
    x[row * KDIM + n] = acc;
}

// ---------------- per-class statistics (deterministic fixed-owner partials) ----------------
// cell 0..9: counts; 10..169: sums[c][n]; 170..2729: Sraw[c][a][b]
__global__ void stats_part_kernel(const float* __restrict__ x, const int* __restrict__ lab,
                                  float* __restrict__ part) {
    int gid = blockIdx.x * blockDim.x + threadIdx.x;
    if (gid >= NCHUNK * NCELL) return;
    int chunk = gid / NCELL;
    int cell  = gid % NCELL;
    int r0 = chunk * 1024;
    float acc = 0.f;
    if (cell < 10) {
        int c = cell;
        for (int r = r0; r < r0 + 1024; ++r) acc += (lab[r] == c) ? 1.f : 0.f;
    } else if (cell < 170) {
        int c = (cell - 10) >> 4, n = (cell - 10) & 15;
        for (int r = r0; r < r0 + 1024; ++r)
            if (lab[r] == c) acc += x[r * KDIM + n];
    } else {
        int q = cell - 170;
        int c = q >> 8, a = (q >> 4) & 15, bb = q & 15;
        for (int r = r0; r < r0 + 1024; ++r)
            if (lab[r] == c) acc += x[r * KDIM + a] * x[r * KDIM + bb];
    }
    part[chunk * NCELL + cell] = acc;
}

__global__ void stats_reduce_kernel(const float* __restrict__ part, float* __restrict__ st) {
    int cell = blockIdx.x * blockDim.x + threadIdx.x;
    if (cell < NCELL) {
        float a = 0.f;
        for (int c = 0; c < NCHUNK; ++c) a += part[c * NCELL + cell];
        st[cell] = a;
    }
}

// ---------------- finalize: Sigma, Cholesky inverse, logdet, pairwise KL ----------------
__global__ void finalize_kernel(const float* __restrict__ st, float* __restrict__ out) {
    __shared__ float cnt[NCLS];
    __shared__ float mean[NCLS][16];
    __shared__ float Sig[NCLS][16][16];
    __shared__ float R[NCLS][16][16];
    __shared__ float Inv[NCLS][16][16];
    __shared__ float logdet[NCLS];
    __shared__ float pr[100];
    int t = threadIdx.x;   // 256
    if (t < NCLS) cnt[t] = st[t];
    __syncthreads();
    if (t < 160) {
        int c = t >> 4, n = t & 15;
        float safe = fmaxf(cnt[c], 1.f);
        mean[c][n] = st[10 + t] / safe;
    }
    __syncthreads();
    for (int idx = t; idx < 2560; idx += 256) {
        int c = idx >> 8, a = (idx >> 4) & 15, b = idx & 15;
        float safe = fmaxf(cnt[c], 1.f);
        float ssum = st[170 + idx] - cnt[c] * mean[c][a] * mean[c][b];  // centered 2nd moment
        Sig[c][a][b] = ((a == b) ? 1.0f : 0.0f) + ssum / safe;          // SIGMA = 1.0
    }
    __syncthreads();
    if (t < NCLS) {
        int c = t;
        for (int i = 0; i < 16; ++i) {                 // Sig = R^T R, R upper
            float d = Sig[c][i][i];
            for (int k = 0; k < i; ++k) d -= R[c][k][i] * R[c][k][i];
            float rii = sqrtf(fmaxf(d, 1e-30f));
            R[c][i][i] = rii;
            float inv = 1.0f / rii;
            for (int j = i + 1; j < 16; ++j) {
                float v = Sig[c][i][j];
                for (int k = 0; k < i; ++k) v -= R[c][k][i] * R[c][k][j];
                R[c][i][j] = v * inv;
            }
        }
        float ld = 0.f;
        for (int i = 0; i < 16; ++i) ld += logf(R[c][i][i]);
        logdet[c] = 2.f * ld;
        float X[16][16];                               // X = R^-1
        for (int j = 0; j < 16; ++j) {
            for (int i = j; i >= 0; --i) {
                float v = (i == j) ? 1.f : 0.f;
                for (int k = i + 1; k <= j; ++k) v -= R[c][i][k] * X[k][j];
                X[i][j] = v / R[c][i][i];
            }
            for (int i = j + 1; i < 16; ++i) X[i][j] = 0.f;
        }
        for (int a = 0; a < 16; ++a)                   // Inv = X X^T
            for (int b = 0; b < 16; ++b) {
                float v = 0.f;
                for (int k = 0; k < 16; ++k) v += X[a][k] * X[b][k];
                Inv[c][a][b] = v;
            }
    }
    __syncthreads();
    float p = 0.f;
    if (t < 100) {
        int i = t / 10, j = t % 10;
        if (i != j && cnt[i] > 0.f && cnt[j] > 0.f) {
            float tr = 0.f;
            for (int a = 0; a < 16; ++a)
                for (int b = 0; b < 16; ++b) tr += Inv[j][a][b] * Sig[i][b][a];
            float maha = 0.f;
            for (int a = 0; a < 16; ++a) {
                float da = mean[j][a] - mean[i][a];
                float acc = 0.f;
                for (int b = 0; b < 16; ++b) acc += Inv[j][a][b] * (mean[j][b] - mean[i][b]);
                maha += da * acc;
            }
            float kl = 0.5f * (tr + maha - 16.f + logdet[j] - logdet[i]);
            p = kl * cnt[i] * cnt[j];
        }
    }
    if (t < 100) pr[t] = p;
    __syncthreads();
    if (t == 0) {
        float tot = 0.f;
        for (int q = 0; q < 100; ++q) tot += pr[q];   // fixed order -> deterministic
        const float B3 = 16384.f * 16384.f * 16384.f; // 2^42, exact in fp32
        out[0] = tot / B3;
    }
}

// ---------------- host-side orchestration ----------------
extern "C" void kernel_launch(void* const* d_in, const int* in_sizes, int n_in,
                              void* d_out, int out_size, void* d_ws, size_t ws_size,
                              hipStream_t stream) {
    const float* mu  = (const float*)d_in[0];   // [16384, 1024] f32
    const int*   lab = (const int*)d_in[1];     // [16384] i32
    float* out = (float*)d_out;                 // scalar f32
    float* ws  = (float*)d_ws;

    float* s     = ws;                          // 1024          column sums
    float* spart = s + 1024;                    // 16*1024       colsum partials
    float* Cm    = spart + 16 * 1024;           // 1024*1024     Gram / centered C
    float* Q     = Cm + 1024 * 1024;            // 1024*16
    float* Z     = Q + 1024 * KDIM;             // 1024*16
    float* r     = Z + 1024 * KDIM;             // 16
    float* x     = r + 16;                      // 16384*16      projected features
    float* stp   = x + (size_t)NROWS * KDIM;    // 16*2730       stats partials
    float* st    = stp + NCHUNK * NCELL;        // 2730          reduced stats

    colsum_part_kernel<<<64, 256, 0, stream>>>(mu, spart);
    colsum_reduce_kernel<<<4, 256, 0, stream>>>(spart, s);

    gram_wmma_kernel<<<1024, 32, 0, stream>>>(mu, Cm);   // 32x32 super-tiles
    center_kernel<<<4096, 256, 0, stream>>>(Cm, s);

    initq_kernel<<<64, 256, 0, stream>>>(Q);
    ortho_kernel<<<1, 256, 0, stream>>>(Q, Q);
    for (int it = 0; it < NCYCLE; ++it) {       // C^2 per ortho cycle
        spmm_kernel<<<64, 256, 0, stream>>>(Cm, Q, Z);
        spmm_kernel<<<64, 256, 0, stream>>>(Cm, Z, Q);
        ortho_kernel<<<1, 256, 0, stream>>>(Q, Q);
    }

    rvec_kernel<<<1, 32, 0, stream>>>(s, Q, r);
    project_kernel<<<1024, 256, 0, stream>>>(mu, Q, r, x);

    stats_part_kernel<<<171, 256, 0, stream>>>(x, lab, stp);
    stats_reduce_kernel<<<11, 256, 0, stream>>>(stp, st);
    finalize_kernel<<<1, 256, 0, stream>>>(st, out);
}